// GCN_83519934038393
// MI455X (gfx1250) — compile-verified
//
#include <hip/hip_runtime.h>
#include <hip/hip_bf16.h>

// ---------------------------------------------------------------------------
// GraphSAGE on MI455X (gfx1250, wave32).
// Dense GEMMs via V_WMMA_F32_16X16X4_F32 (exact fp32).
// Aggregation via CSR gather (built per call with counting sort) -> streaming,
// no float atomics in the hot loops; h/agg are L2-resident (51MB each, 192MB L2).
// ---------------------------------------------------------------------------

#define N_NODES 100000
#define N_EDGES 1600000
#define IN_C    100
#define HID     128
#define OUT_C   47
#define OUT_P   48   // padded output width for WMMA N-tiles

typedef __attribute__((ext_vector_type(2))) float v2f;
typedef __attribute__((ext_vector_type(8))) float v8f;

__device__ __forceinline__ v8f wmma4(v2f a, v2f b, v8f c) {
    // D = A(16x4) * B(4x16) + C(16x16), fp32 exact
    return __builtin_amdgcn_wmma_f32_16x16x4_f32(false, a, false, b,
                                                 (short)0, c, false, false);
}

// ---------------- degree / CSR build -----------------------------------------

__global__ void k_deg(const int* __restrict__ dst, int* __restrict__ deg) {
    int e = blockIdx.x * 256 + threadIdx.x;
    if (e < N_EDGES) atomicAdd(&deg[dst[e]], 1);
}

// single-block exclusive scan over 100000 degrees -> row_start/cursor/inv_deg
__global__ __launch_bounds__(1024) void k_scan(const int* __restrict__ deg,
                                               int* __restrict__ row_start,
                                               int* __restrict__ cursor,
                                               float* __restrict__ inv_deg) {
    __shared__ int s[1024];
    const int n = N_NODES;
    const int t = threadIdx.x;
    const int seg = (n + 1023) >> 10;            // 98
    const int lo = t * seg;
    const int hi = (lo + seg < n) ? lo + seg : n;
    int sum = 0;
    for (int i = lo; i < hi; ++i) sum += deg[i];
    s[t] = sum;
    __syncthreads();
    for (int off = 1; off < 1024; off <<= 1) {   // inclusive Hillis-Steele
        int v = s[t];
        int add = (t >= off) ? s[t - off] : 0;
        __syncthreads();
        s[t] = v + add;
        __syncthreads();
    }
    int run = (t == 0) ? 0 : s[t - 1];
    for (int i = lo; i < hi; ++i) {
        row_start[i] = run;
        cursor[i]    = run;
        int d = deg[i];
        inv_deg[i] = 1.0f / (float)(d > 1 ? d : 1);
        run += d;
    }
    if (t == 1023) row_start[n] = s[1023];
}

__global__ void k_fill(const int* __restrict__ src, const int* __restrict__ dst,
                       int* __restrict__ cursor, int* __restrict__ csr) {
    int e = blockIdx.x * 256 + threadIdx.x;
    if (e < N_EDGES) {
        int d = dst[e];
        int p = atomicAdd(&cursor[d], 1);
        csr[p] = src[e];
    }
}

// ---------------- mean aggregation (CSR gather) ------------------------------

__global__ __launch_bounds__(128) void k_agg(const float* __restrict__ h,
                                             const int* __restrict__ row_start,
                                             const int* __restrict__ csr,
                                             const float* __restrict__ inv_deg,
                                             float* __restrict__ agg) {
    const int node = blockIdx.x;
    const int c = threadIdx.x;                 // 128 channels
    const int s0 = row_start[node];
    const int s1 = row_start[node + 1];
    float sum = 0.0f;
    for (int j = s0; j < s1; ++j) {
        int sidx = csr[j];                     // uniform per block
        sum += h[(long)sidx * HID + c];        // 512B coalesced row read
    }
    agg[(long)node * HID + c] = sum * inv_deg[node];
}

// ---------------- input projection GEMM (K=100, N=128) -----------------------
// inp = X @ Wi + bi   ;   h = relu(inp)

__global__ __launch_bounds__(256) void k_gemm_in(const float* __restrict__ X,
                                                 const float* __restrict__ W,
                                                 const float* __restrict__ bias,
                                                 float* __restrict__ inp,
                                                 float* __restrict__ h) {
    __shared__ float xs[16 * IN_C];            // stride 100 (no bank conflicts)
    const int base = blockIdx.x * 16;
    const int tid  = threadIdx.x;
    // 16 rows x 100 floats contiguous in global -> straight float4 copy
    const float4* xg = (const float4*)(X + (long)base * IN_C);
    float4* xl = (float4*)xs;
    for (int i = tid; i < (16 * IN_C) / 4; i += 256) xl[i] = xg[i];
    __syncthreads();

    const int lane = tid & 31, wid = tid >> 5;
    const int lr  = lane & 15;
    const int hi2 = (lane >> 4) << 1;          // 0 or 2 (K sub-pair)
    const int n0  = wid * 16;

    v8f acc;
    float bv = bias[n0 + lr];
#pragma unroll
    for (int i = 0; i < 8; ++i) acc[i] = bv;

#pragma unroll
    for (int k = 0; k < IN_C; k += 4) {
        float2 af = *(const float2*)&xs[lr * IN_C + k + hi2];   // ds_load_b64
        v2f a; a[0] = af.x; a[1] = af.y;
        v2f b;
        b[0] = W[(k     + hi2) * HID + n0 + lr];
        b[1] = W[(k + 1 + hi2) * HID + n0 + lr];
        acc = wmma4(a, b, acc);
    }

#pragma unroll
    for (int i = 0; i < 8; ++i) {
        int row = base + i + (hi2 << 2);       // +0 or +8
        int col = n0 + lr;
        float v = acc[i];
        inp[(long)row * HID + col] = v;
        h  [(long)row * HID + col] = fmaxf(v, 0.0f);
    }
}

// ---------------- SAGE layer GEMM: out = A1@W1 + A2@W2 + bias ----------------
// hidden: out = relu(acc) + 0.2*inp ; final (inp==null): out = acc (47 cols)

__global__ __launch_bounds__(256) void k_gemm_sage(const float* __restrict__ A1,
                                                   const float* __restrict__ W1,
                                                   const float* __restrict__ A2,
                                                   const float* __restrict__ W2,
                                                   const float* __restrict__ bias,
                                                   const float* __restrict__ inp,
                                                   float* __restrict__ out,
                                                   int ntile, int ncols,
                                                   int out_ld, int w_ld) {
    __shared__ float a1s[16 * 132];            // stride 132: 8B-aligned, conflict-free
    __shared__ float a2s[16 * 132];
    const int base = blockIdx.x * 16;
    const int tid  = threadIdx.x;

    const float4* g1 = (const float4*)(A1 + (long)base * HID);
    const float4* g2 = (const float4*)(A2 + (long)base * HID);
    for (int i = tid; i < 512; i += 256) {     // 16x128 floats = 512 float4 each
        int r = i >> 5, c4 = i & 31;
        *(float4*)&a1s[r * 132 + c4 * 4] = g1[i];
        *(float4*)&a2s[r * 132 + c4 * 4] = g2[i];
    }
    __syncthreads();

    const int lane = tid & 31, wid = tid >> 5;
    if (wid < ntile) {
        const int lr  = lane & 15;
        const int hi2 = (lane >> 4) << 1;
        const int n0  = wid * 16;

        v8f acc;
        float bv = bias[n0 + lr];
#pragma unroll
        for (int i = 0; i < 8; ++i) acc[i] = bv;

#pragma unroll 4
        for (int k = 0; k < HID; k += 4) {
            float2 af = *(const float2*)&a1s[lr * 132 + k + hi2];
            v2f a; a[0] = af.x; a[1] = af.y;
            v2f b;
            b[0] = W1[(k     + hi2) * w_ld + n0 + lr];
            b[1] = W1[(k + 1 + hi2) * w_ld + n0 + lr];
            acc = wmma4(a, b, acc);

            af = *(const float2*)&a2s[lr * 132 + k + hi2];
            a[0] = af.x; a[1] = af.y;
            b[0] = W2[(k     + hi2) * w_ld + n0 + lr];
            b[1] = W2[(k + 1 + hi2) * w_ld + n0 + lr];
            acc = wmma4(a, b, acc);
        }

#pragma unroll
        for (int i = 0; i < 8; ++i) {
            int row = base + i + (hi2 << 2);
            int col = n0 + lr;
            if (col < ncols) {
                float v = acc[i];
                if (inp) v = fmaxf(v, 0.0f) + 0.2f * inp[(long)row * HID + col];
                out[(long)row * out_ld + col] = v;
            }
        }
    }
}

// ---------------- weight padding (47 -> 48 cols) -----------------------------

__global__ void k_pad_w(const float* __restrict__ w, float* __restrict__ wp) {
    int i = blockIdx.x * 256 + threadIdx.x;    // 128*48
    if (i < HID * OUT_P) {
        int r = i / OUT_P, c = i - r * OUT_P;
        wp[i] = (c < OUT_C) ? w[r * OUT_C + c] : 0.0f;
    }
}
__global__ void k_pad_b(const float* __restrict__ b, float* __restrict__ bp) {
    int i = threadIdx.x;
    if (i < OUT_P) bp[i] = (i < OUT_C) ? b[i] : 0.0f;
}

// ---------------- log-softmax over 47 classes, one wave32 per row ------------

__global__ __launch_bounds__(256) void k_lsm(float* __restrict__ out) {
    const int wid = threadIdx.x >> 5, lane = threadIdx.x & 31;
    const int row = blockIdx.x * 8 + wid;
    if (row >= N_NODES) return;
    float* p = out + (long)row * OUT_C;
    float v0 = p[lane];                                   // lanes 0..31 (<47)
    float v1 = (lane < OUT_C - 32) ? p[32 + lane] : -3.0e38f;
    float m = fmaxf(v0, v1);
#pragma unroll
    for (int off = 16; off; off >>= 1) m = fmaxf(m, __shfl_xor(m, off, 32));
    float s = expf(v0 - m) + ((lane < OUT_C - 32) ? expf(v1 - m) : 0.0f);
#pragma unroll
    for (int off = 16; off; off >>= 1) s += __shfl_xor(s, off, 32);
    float l = m + logf(s);
    p[lane] = v0 - l;
    if (lane < OUT_C - 32) p[32 + lane] = v1 - l;
}

// ---------------- host: orchestrate on `stream` ------------------------------

extern "C" void kernel_launch(void* const* d_in, const int* in_sizes, int n_in,
                              void* d_out, int out_size, void* d_ws, size_t ws_size,
                              hipStream_t stream) {
    const float* x    = (const float*)d_in[0];
    const int*   ei   = (const int*)  d_in[1];
    const float* wi   = (const float*)d_in[2];
    const float* bi   = (const float*)d_in[3];
    const float* wl_h = (const float*)d_in[4];   // 3 x 128 x 128
    const float* bl_h = (const float*)d_in[5];   // 3 x 128
    const float* wr_h = (const float*)d_in[6];   // 3 x 128 x 128
    const float* wl_o = (const float*)d_in[7];   // 128 x 47
    const float* bl_o = (const float*)d_in[8];   // 47
    const float* wr_o = (const float*)d_in[9];   // 128 x 47
    const int* src = ei;
    const int* dst = ei + N_EDGES;

    char* w = (char*)d_ws;
    auto alloc = [&](size_t bytes) -> char* {
        char* p = w;
        w += (bytes + 255) & ~(size_t)255;
        return p;
    };
    const size_t feat_bytes = (size_t)N_NODES * HID * sizeof(float);
    float* inp      = (float*)alloc(feat_bytes);
    float* hA       = (float*)alloc(feat_bytes);
    float* hB       = (float*)alloc(feat_bytes);
    float* agg      = (float*)alloc(feat_bytes);
    float* inv_deg  = (float*)alloc((size_t)N_NODES * 4);
    float* wlp      = (float*)alloc((size_t)HID * OUT_P * 4);
    float* wrp      = (float*)alloc((size_t)HID * OUT_P * 4);
    float* blp      = (float*)alloc((size_t)OUT_P * 4);
    int*   deg      = (int*)  alloc((size_t)N_NODES * 4);
    int*   rowstart = (int*)  alloc((size_t)(N_NODES + 1) * 4);
    int*   cursor   = (int*)  alloc((size_t)N_NODES * 4);
    int*   csr      = (int*)  alloc((size_t)N_EDGES * 4);

    const int EB = (N_EDGES + 255) / 256;      // 6250
    const int MB = N_NODES / 16;               // 6250

    // CSR build (counting sort by dst)
    hipMemsetAsync(deg, 0, (size_t)N_NODES * 4, stream);
    k_deg <<<EB, 256, 0, stream>>>(dst, deg);
    k_scan<<<1, 1024, 0, stream>>>(deg, rowstart, cursor, inv_deg);
    k_fill<<<EB, 256, 0, stream>>>(src, dst, cursor, csr);

    // pad output-layer weights to 48 cols
    k_pad_w<<<(HID * OUT_P + 255) / 256, 256, 0, stream>>>(wl_o, wlp);
    k_pad_w<<<(HID * OUT_P + 255) / 256, 256, 0, stream>>>(wr_o, wrp);
    k_pad_b<<<1, 64, 0, stream>>>(bl_o, blp);

    // input projection: inp = x@Wi + bi ; hA = relu(inp)
    k_gemm_in<<<MB, 256, 0, stream>>>(x, wi, bi, inp, hA);

    float* hc = hA;
    float* hn = hB;
    for (int L = 0; L < 3; ++L) {
        k_agg<<<N_NODES, 128, 0, stream>>>(hc, rowstart, csr, inv_deg, agg);
        k_gemm_sage<<<MB, 256, 0, stream>>>(agg, wl_h + (size_t)L * HID * HID,
                                            hc,  wr_h + (size_t)L * HID * HID,
                                            bl_h + (size_t)L * HID, inp, hn,
                                            8, HID, HID, HID);
        float* t = hc; hc = hn; hn = t;
    }

    // final SAGE -> logits (47 cols), then in-place log-softmax
    k_agg<<<N_NODES, 128, 0, stream>>>(hc, rowstart, csr, inv_deg, agg);
    k_gemm_sage<<<MB, 256, 0, stream>>>(agg, wlp, hc, wrp, blp, nullptr,
                                        (float*)d_out, 3, OUT_C, OUT_C, OUT_P);
    k_lsm<<<(N_NODES + 7) / 8, 256, 0, stream>>>((float*)d_out);
}